// FractalExpert_84705345011960
// MI455X (gfx1250) — compile-verified
//
#include <hip/hip_runtime.h>

#define NN 15          // N_NODES
#define NI 7           // N_INTERNAL
#define DD 4096        // feature dim
#define BB 8192        // batch

typedef __attribute__((ext_vector_type(2))) float v2f;
typedef __attribute__((ext_vector_type(8))) float v8f;

// ---- tree coefficients (uniform across all threads; tiny fixed loop) ----
__device__ __forceinline__ void compute_c(const float* __restrict__ rand_vals,
                                          const float* __restrict__ depth_prob,
                                          float* c) {
  float coef[NN];
#pragma unroll
  for (int i = 0; i < NN; ++i) coef[i] = 0.0f;
  coef[0] = 1.0f;
  const float dp = depth_prob[0];
#pragma unroll
  for (int i = 0; i < NI; ++i) {
    const float d = (rand_vals[i] < dp) ? 1.0f : 0.0f;
    c[i] = coef[i] * (1.0f - d);
    const float half = coef[i] * d * 0.5f;
    coef[2 * i + 1] += half;
    coef[2 * i + 2] += half;
  }
#pragma unroll
  for (int i = NI; i < NN; ++i) c[i] = coef[i];
}

// ---- kernel 1: wc[d] = sum_n c_n * W[n,d];  bc[d] = sum_n c_n * bias[n,d] ----
__global__ void fe_combine_kernel(const float* __restrict__ weights,
                                  const float* __restrict__ bias,
                                  const float* __restrict__ rand_vals,
                                  const float* __restrict__ depth_prob,
                                  float* __restrict__ wc,
                                  float* __restrict__ bc) {
  const int d = blockIdx.x * blockDim.x + threadIdx.x;
  if (d >= DD) return;
  float c[NN];
  compute_c(rand_vals, depth_prob, c);
  float w = 0.0f, bsum = 0.0f;
#pragma unroll
  for (int n = 0; n < NN; ++n) {
    w += c[n] * weights[n * DD + d];
    bsum += c[n] * bias[n * DD + d];
  }
  wc[d] = w;
  bc[d] = bsum;
}

// ---- kernel 2: sdot[b] = x[b] . wc via V_WMMA_F32_16X16X4_F32 ----
// One block = 4 waves = one 16-row tile; split-K (1024 floats per wave).
// A-matrix layout (16x4 f32): lane m in [0,16) holds row m; VGPR0 = K(0|2),
// VGPR1 = K(1|3), upper half-wave carries K=2,3 -> per-lane b64 load at
// x[row*D + k + 2*hi]. B-matrix = wc broadcast to all 16 columns, so every
// column of the 16x16 accumulator equals the row dot product.
__global__ void __launch_bounds__(128) fe_dot_kernel(const float* __restrict__ x,
                                                     const float* __restrict__ wc,
                                                     float* __restrict__ sdot) {
  __shared__ float part[64];

  const int lane = threadIdx.x & 31;
  const int wave = threadIdx.x >> 5;
  const int r0   = blockIdx.x * 16;
  const int m    = lane & 15;
  const int hi   = lane >> 4;

  const float* __restrict__ xrow = x + (size_t)(r0 + m) * DD + 2 * hi;
  const float* __restrict__ wcp  = wc + 2 * hi;
  const int kbase = wave * (DD / 4);   // 1024 floats per wave

  v8f acc = {};
#pragma unroll 8
  for (int k = kbase; k < kbase + DD / 4; k += 4) {
    v2f a = *(const v2f*)(xrow + k);   // A: (M=m, K=k+2hi, k+2hi+1)
    v2f b = *(const v2f*)(wcp + k);    // B: row K broadcast across N columns
    acc = __builtin_amdgcn_wmma_f32_16x16x4_f32(
        /*neg_a=*/false, a, /*neg_b=*/false, b,
        /*c_mod=*/(short)0, acc, /*reuse_a=*/false, /*reuse_b=*/false);
  }

  // C layout: lane n<16 VGPR j -> (M=j, N=n); lane n+16 -> (M=j+8, N=n).
  // Columns are identical, so lane 0 holds rows r0..r0+7, lane 16 rows r0+8..r0+15.
  if ((lane & 15) == 0) {
#pragma unroll
    for (int j = 0; j < 8; ++j)
      part[wave * 16 + hi * 8 + j] = acc[j];
  }
  __syncthreads();

  if (threadIdx.x < 16) {
    const float s = part[threadIdx.x] + part[16 + threadIdx.x] +
                    part[32 + threadIdx.x] + part[48 + threadIdx.x];
    sdot[r0 + threadIdx.x] = s;
  }
}

// ---- kernel 3: out[b,o] = sdot[b] + bc[o], one row per block, float4 stores ----
__global__ void __launch_bounds__(1024) fe_bcast_kernel(const float* __restrict__ sdot,
                                                        const float* __restrict__ bc,
                                                        float* __restrict__ out) {
  const int b = blockIdx.x;
  const float s = sdot[b];
  const int o = threadIdx.x;                       // float4 index, 0..1023
  const float4 bv = ((const float4*)bc)[o];
  float4 r;
  r.x = s + bv.x;
  r.y = s + bv.y;
  r.z = s + bv.z;
  r.w = s + bv.w;
  ((float4*)out)[(size_t)b * (DD / 4) + o] = r;
}

extern "C" void kernel_launch(void* const* d_in, const int* in_sizes, int n_in,
                              void* d_out, int out_size, void* d_ws, size_t ws_size,
                              hipStream_t stream) {
  const float* x          = (const float*)d_in[0];  // (B, D)
  const float* weights    = (const float*)d_in[1];  // (15, D)
  const float* bias       = (const float*)d_in[2];  // (15, D)
  const float* rand_vals  = (const float*)d_in[3];  // (7,)
  const float* depth_prob = (const float*)d_in[4];  // (1,)
  float* out = (float*)d_out;                       // (B, D)

  float* wc   = (float*)d_ws;          // D floats
  float* bc   = wc + DD;               // D floats
  float* sdot = bc + DD;               // B floats

  // 1) fold tree into wc / bc vectors
  fe_combine_kernel<<<DD / 256, 256, 0, stream>>>(weights, bias, rand_vals,
                                                  depth_prob, wc, bc);
  // 2) per-row dot products via WMMA (reads x: 134 MB)
  fe_dot_kernel<<<BB / 16, 128, 0, stream>>>(x, wc, sdot);
  // 3) rank-1 broadcast write (writes out: 134 MB)
  fe_bcast_kernel<<<BB, 1024, 0, stream>>>(sdot, bc, out);
}